// DeterministicAdjacency_42082089566385
// MI455X (gfx1250) — compile-verified
//
#include <hip/hip_runtime.h>

// ---------------------------------------------------------------------------
// DeterministicAdjacency for MI455X (gfx1250, wave32).
//
//   hi = z @ W1[:128] + b1   (fold bias)        -> Hcat[:, 0:64]
//   hj = z @ W1[128:]                           -> Hcat[:, 64:128]
//   logits[i,j] = sum_e silu(hi[i,e]+hj[j,e]) * w2[e]   (+b2 dropped: softmax
//                                                        is shift-invariant)
//   out = softmax_j(logits)
//
// Kernel A: f32 WMMA GEMM (V_WMMA_F32_16X16X4_F32), one 16x16 tile per wave.
// Kernel B: fused silu-dot + row softmax, TI=4 rows/block to amortize hj
//           reads (L2-resident, 192MB L2 >> 1MB Hcat).
// ---------------------------------------------------------------------------

typedef float v2f __attribute__((ext_vector_type(2)));
typedef float v8f __attribute__((ext_vector_type(8)));

#define KROWS 2048
#define DDIM  128
#define EDIM  64
#define TI    4

// ---------------- Kernel A: Hcat = z @ [W1_lo | W1_hi], +b1 on lo half -----
__global__ __launch_bounds__(32) void adj_gemm_kernel(
    const float* __restrict__ Z,    // (2048,128)
    const float* __restrict__ W1,   // (256,64) row-major
    const float* __restrict__ b1,   // (64)
    float* __restrict__ H)          // (2048,128) workspace
{
    const int lane  = threadIdx.x;        // 0..31
    const int lm    = lane & 15;          // A row / B col / D col
    const int half  = lane >> 4;          // 0/1: K-halves for A/B, M-halves for D
    const int tileM = blockIdx.x * 16;    // 128 tiles of M
    const int tileN = blockIdx.y * 16;    // 8 tiles of N (cols of Hcat)

    const bool isHi = (tileN < 64);
    const int  ncol = tileN + lm;
    // W'[k][n] = (n<64) ? W1[k][n] : W1[128+k][n-64];  row stride 64 floats
    const float* wcol = isHi ? (W1 + ncol) : (W1 + 128 * 64 + (ncol - 64));
    const float* zrow = Z + (size_t)(tileM + lm) * DDIM;

    v8f c = {};
    for (int k0 = 0; k0 < DDIM; k0 += 4) {
        const int ka = k0 + 2 * half;     // this lane-half's K pair
        // A fragment: lanes 0-15 hold K=k0,k0+1; lanes 16-31 hold K=k0+2,k0+3
        v2f a = *(const v2f*)(zrow + ka); // 8B-aligned (ka even)
        // B fragment: VGPR0 -> K=ka, VGPR1 -> K=ka+1, col = lane&15
        v2f b;
        b.x = wcol[(size_t)ka * EDIM];
        b.y = wcol[(size_t)(ka + 1) * EDIM];
        c = __builtin_amdgcn_wmma_f32_16x16x4_f32(
                /*neg_a=*/false, a, /*neg_b=*/false, b,
                /*c_mod=*/(short)0, c, /*reuse_a=*/false, /*reuse_b=*/false);
    }

    // D layout: VGPR r -> M = r + 8*half, N = lane&15
    const float bias = isHi ? b1[ncol] : 0.0f;
#pragma unroll
    for (int r = 0; r < 8; ++r) {
        const int row = tileM + r + 8 * half;
        H[(size_t)row * DDIM + tileN + lm] = c[r] + bias;
    }
}

// ---------------- Kernel B: fused silu-dot + softmax -----------------------
__global__ __launch_bounds__(256) void adj_softmax_kernel(
    const float* __restrict__ H,    // (2048,128): [hi+b1 | hj]
    const float* __restrict__ W2,   // (64)
    float* __restrict__ out)        // (2048,2048)
{
    __shared__ float sHi[TI][EDIM];     // 1 KB
    __shared__ float sW2[EDIM];         // 256 B
    __shared__ float sLog[TI][KROWS];   // 32 KB
    __shared__ float sInv[TI];

    const int t  = threadIdx.x;         // 0..255 (8 waves)
    const int i0 = blockIdx.x * TI;
    const float LOG2E = 1.4426950408889634f;

    if (t < EDIM) sW2[t] = W2[t];
    for (int idx = t; idx < TI * EDIM; idx += 256) {
        const int r = idx >> 6, e = idx & 63;
        sHi[r][e] = H[(size_t)(i0 + r) * DDIM + e];
    }
    __syncthreads();

    // ---- Phase 1: logits for TI rows, each thread owns 8 columns j --------
    for (int jj = 0; jj < KROWS / 256; ++jj) {
        const int j = jj * 256 + t;
        // hj row into registers (16 x float4)
        float hj[EDIM];
        const float4* hp = (const float4*)(H + (size_t)j * DDIM + EDIM);
#pragma unroll
        for (int q = 0; q < EDIM / 4; ++q) {
            const float4 v = hp[q];
            hj[4 * q + 0] = v.x; hj[4 * q + 1] = v.y;
            hj[4 * q + 2] = v.z; hj[4 * q + 3] = v.w;
        }
        float acc[TI] = {0.f, 0.f, 0.f, 0.f};
#pragma unroll
        for (int e = 0; e < EDIM; ++e) {          // fully unrolled: hj in VGPRs
            const float hv = hj[e];
            const float w  = sW2[e];
#pragma unroll
            for (int r = 0; r < TI; ++r) {
                const float x  = sHi[r][e] + hv;
                // silu(x) = x / (1 + exp(-x)); v_exp_f32 + v_rcp_f32 fast path
                const float ex = __builtin_amdgcn_exp2f(-x * LOG2E);
                const float s  = x * __builtin_amdgcn_rcpf(1.0f + ex);
                acc[r] = fmaf(s, w, acc[r]);
            }
        }
#pragma unroll
        for (int r = 0; r < TI; ++r) sLog[r][j] = acc[r];
    }
    __syncthreads();

    // ---- Phase 2: per-row max + exp-sum; wave w handles row w -------------
    const int wave = t >> 5, lane = t & 31;
    if (wave < TI) {
        const int r = wave;
        float mx = -3.402823466e38f;
        for (int idx = lane; idx < KROWS; idx += 32)
            mx = fmaxf(mx, sLog[r][idx]);
#pragma unroll
        for (int off = 16; off >= 1; off >>= 1)
            mx = fmaxf(mx, __shfl_xor(mx, off, 32));
        float sum = 0.0f;
        for (int idx = lane; idx < KROWS; idx += 32) {
            const float ev = __builtin_amdgcn_exp2f((sLog[r][idx] - mx) * LOG2E);
            sLog[r][idx] = ev;
            sum += ev;
        }
#pragma unroll
        for (int off = 16; off >= 1; off >>= 1)
            sum += __shfl_xor(sum, off, 32);
        if (lane == 0) sInv[r] = __builtin_amdgcn_rcpf(sum);
    }
    __syncthreads();

    // ---- Phase 3: normalize + store (coalesced) ---------------------------
#pragma unroll
    for (int r = 0; r < TI; ++r) {
        const float inv = sInv[r];
        float* orow = out + (size_t)(i0 + r) * KROWS;
        for (int jj = 0; jj < KROWS / 256; ++jj) {
            const int j = jj * 256 + t;
            orow[j] = sLog[r][j] * inv;
        }
    }
}

// ---------------------------------------------------------------------------
extern "C" void kernel_launch(void* const* d_in, const int* in_sizes, int n_in,
                              void* d_out, int out_size, void* d_ws, size_t ws_size,
                              hipStream_t stream) {
    const float* z  = (const float*)d_in[0];   // (2048,128)
    const float* W1 = (const float*)d_in[1];   // (256,64)
    const float* b1 = (const float*)d_in[2];   // (64)
    const float* W2 = (const float*)d_in[3];   // (64,1)
    // d_in[4] = b2: constant logit shift, cancels in softmax — unused.
    float* out = (float*)d_out;                // (2048,2048)
    float* H   = (float*)d_ws;                 // 2048*128 f32 = 1 MB scratch

    dim3 gA(KROWS / 16, DDIM / 16);            // 128 x 8 tiles, 1 wave each
    adj_gemm_kernel<<<gA, 32, 0, stream>>>(z, W1, b1, H);

    adj_softmax_kernel<<<KROWS / TI, 256, 0, stream>>>(H, W2, out);
}